// GroupWeight_49898930045024
// MI455X (gfx1250) — compile-verified
//
#include <hip/hip_runtime.h>
#include <hip/hip_bf16.h>

#define SELF_DIM 36
#define ENT_DIM  28
#define HID      256
#define OBS_DIM  1688   // 36 + 28*59
#define NSAMPLES 32768

typedef float v2f __attribute__((ext_vector_type(2)));
typedef float v8f __attribute__((ext_vector_type(8)));

// Workspace layout (floats):
//   Mfrag : 16 N-tiles * 16 K-steps * 32 lanes * 2 = 16384   (folded 64x256 matrix in WMMA-B fragment order)
//   cvec  : 256                                              (folded output bias)
//   G     : 36*3 = 108                                       (self -> gate logits)
//   H     : 3*28*3 = 252                                     (p_g -> gate logits, per group)
//   cg    : 3                                                (gate bias)
#define WS_MFRAG 0
#define WS_CVEC  16384
#define WS_G     16640
#define WS_H     16748
#define WS_CG    17000
#define WS_TOTAL 17003

__global__ __launch_bounds__(256) void fold_weights_kernel(
    const float* __restrict__ We,  const float* __restrict__ be,
    const float* __restrict__ Wse, const float* __restrict__ bse,
    const float* __restrict__ Wg,  const float* __restrict__ bg,
    const float* __restrict__ Wo,  const float* __restrict__ bo,
    float* __restrict__ ws) {
  int tid = blockIdx.x * blockDim.x + threadIdx.x;
  if (tid < 16384) {
    // M[i][n], i in [0,64), n in [0,256)
    int i = tid >> 8;
    int n = tid & 255;
    float acc = 0.f;
    if (i < SELF_DIM) {
      for (int k = 0; k < HID; ++k)
        acc += We[i * HID + k] * Wo[k * HID + n];
      for (int k = 0; k < HID; ++k)
        acc += (Wse[i * HID + k] + Wse[(i + SELF_DIM) * HID + k]) * Wo[(HID + k) * HID + n];
    } else {
      for (int k = 0; k < HID; ++k)
        acc += We[i * HID + k] * Wo[k * HID + n];
    }
    // scatter into B-fragment order:
    // frag f = t*16+kk; lane holds pair (row 4kk+c, 4kk+c+1), col 16t+(lane&15), c=2*(lane>=16)
    int t = n >> 4, lanelo = n & 15;
    int kk = i >> 2, rr = i & 3;
    int half = rr >> 1, pidx = rr & 1;
    int lane = lanelo + 16 * half;
    ws[WS_MFRAG + (((t * 16) + kk) * 32 + lane) * 2 + pidx] = acc;
  } else if (tid < 16384 + 256) {
    int n = tid - 16384;
    float acc = bo[n];
    for (int k = 0; k < HID; ++k)
      acc += be[k] * Wo[k * HID + n] + bse[k] * Wo[(HID + k) * HID + n];
    ws[WS_CVEC + n] = acc;
  } else if (tid < 16384 + 256 + 108) {
    int r = tid - WS_G;
    int d = r / 3, j = r % 3;
    float acc = 0.f;
    for (int k = 0; k < HID; ++k)
      acc += We[d * HID + k] * (Wg[k * 3 + j] + Wg[(HID + k) * 3 + j] + Wg[(2 * HID + k) * 3 + j]);
    ws[WS_G + r] = acc;
  } else if (tid < 16384 + 256 + 108 + 252) {
    int r = tid - WS_H;
    int g = r / 84, rem = r % 84, d = rem / 3, j = rem % 3;
    float acc = 0.f;
    for (int k = 0; k < HID; ++k)
      acc += We[(SELF_DIM + d) * HID + k] * Wg[(g * HID + k) * 3 + j];
    ws[WS_H + r] = acc;
  } else if (tid < WS_TOTAL) {
    int j = tid - WS_CG;
    float acc = bg[j];
    for (int k = 0; k < HID; ++k)
      acc += be[k] * (Wg[k * 3 + j] + Wg[(HID + k) * 3 + j] + Wg[(2 * HID + k) * 3 + j]);
    ws[WS_CG + j] = acc;
  }
}

// One wave (32 lanes) per block, 32 samples per block.
__global__ __launch_bounds__(32) void obs_embed_kernel(
    const float* __restrict__ x, const float* __restrict__ Watt,
    const float* __restrict__ ws, float* __restrict__ out) {
  __shared__ float lds_p[32 * 3 * ENT_DIM];  // pooled entity vectors
  __shared__ float lds_hl[32 * 3 * 3];       // p_g @ H_g partial gate logits
  __shared__ float lds_v[32 * 64];           // [self(36), p_c(28)] per sample

  const int lane = threadIdx.x;
  const int sample0 = blockIdx.x * 32;

  // attention weight on entity part only (self part + batt cancel in softmax)
  float w2[ENT_DIM];
#pragma unroll
  for (int d = 0; d < ENT_DIM; ++d) w2[d] = Watt[SELF_DIM + d];

  // ---------------- phase 1: per-(sample,group) online-softmax pooling ----------------
  for (int round = 0; round < 3; ++round) {
    int t = round * 32 + lane;          // 96 tasks over 3 full rounds
    int s = t / 3;
    int g = t - 3 * s;
    int off = (g == 0) ? SELF_DIM : ((g == 1) ? (SELF_DIM + 20 * ENT_DIM)
                                              : (SELF_DIM + 39 * ENT_DIM));
    int n = (g == 1) ? 19 : 20;
    const float* base = x + (long long)(sample0 + s) * OBS_DIM + off;

    float m = -__builtin_inff();
    float denom = 0.f;
    float acc[ENT_DIM];
#pragma unroll
    for (int d = 0; d < ENT_DIM; ++d) acc[d] = 0.f;

    for (int e = 0; e < n; ++e) {
      float ent[ENT_DIM];
      const float2* ep = (const float2*)(base + e * ENT_DIM);  // 8B aligned
#pragma unroll
      for (int d2 = 0; d2 < ENT_DIM / 2; ++d2) {
        float2 q = ep[d2];
        ent[2 * d2] = q.x;
        ent[2 * d2 + 1] = q.y;
      }
      float lg = 0.f;
#pragma unroll
      for (int d = 0; d < ENT_DIM; ++d) lg = fmaf(ent[d], w2[d], lg);
      float nm = fmaxf(m, lg);
      float sc = __expf(m - nm);      // first iter: exp(-inf)=0
      float w  = __expf(lg - nm);
      denom = denom * sc + w;
#pragma unroll
      for (int d = 0; d < ENT_DIM; ++d) acc[d] = acc[d] * sc + w * ent[d];
      m = nm;
    }

    float inv = 1.f / denom;
    const float* Hg = ws + WS_H + g * (ENT_DIM * 3);
    float hl0 = 0.f, hl1 = 0.f, hl2 = 0.f;
#pragma unroll
    for (int d = 0; d < ENT_DIM; ++d) {
      float p = acc[d] * inv;
      lds_p[(s * 3 + g) * ENT_DIM + d] = p;
      hl0 = fmaf(p, Hg[d * 3 + 0], hl0);
      hl1 = fmaf(p, Hg[d * 3 + 1], hl1);
      hl2 = fmaf(p, Hg[d * 3 + 2], hl2);
    }
    lds_hl[(s * 3 + g) * 3 + 0] = hl0;
    lds_hl[(s * 3 + g) * 3 + 1] = hl1;
    lds_hl[(s * 3 + g) * 3 + 2] = hl2;
  }
  __syncthreads();

  // ---------------- phase 2: gate softmax + build v = [self, p_c] ----------------
  {
    int s = lane;
    const float* sb = x + (long long)(sample0 + s) * OBS_DIM;
    float selfv[SELF_DIM];
#pragma unroll
    for (int d4 = 0; d4 < SELF_DIM / 4; ++d4) {
      float4 q = ((const float4*)sb)[d4];  // 16B aligned (stride 6752B)
      selfv[4 * d4 + 0] = q.x; selfv[4 * d4 + 1] = q.y;
      selfv[4 * d4 + 2] = q.z; selfv[4 * d4 + 3] = q.w;
    }
    const float* G = ws + WS_G;
    float l0 = ws[WS_CG + 0], l1 = ws[WS_CG + 1], l2 = ws[WS_CG + 2];
#pragma unroll
    for (int d = 0; d < SELF_DIM; ++d) {
      l0 = fmaf(selfv[d], G[d * 3 + 0], l0);
      l1 = fmaf(selfv[d], G[d * 3 + 1], l1);
      l2 = fmaf(selfv[d], G[d * 3 + 2], l2);
    }
#pragma unroll
    for (int g = 0; g < 3; ++g) {
      l0 += lds_hl[(s * 3 + g) * 3 + 0];
      l1 += lds_hl[(s * 3 + g) * 3 + 1];
      l2 += lds_hl[(s * 3 + g) * 3 + 2];
    }
    float mx = fmaxf(l0, fmaxf(l1, l2));
    float e0 = __expf(l0 - mx), e1 = __expf(l1 - mx), e2 = __expf(l2 - mx);
    float is = 1.f / (e0 + e1 + e2);
    float g0 = e0 * is, g1 = e1 * is, g2 = e2 * is;

    float* vrow = lds_v + s * 64;
#pragma unroll
    for (int d = 0; d < SELF_DIM; ++d) vrow[d] = selfv[d];
#pragma unroll
    for (int d = 0; d < ENT_DIM; ++d) {
      vrow[SELF_DIM + d] = g0 * lds_p[(s * 3 + 0) * ENT_DIM + d]
                         + g1 * lds_p[(s * 3 + 1) * ENT_DIM + d]
                         + g2 * lds_p[(s * 3 + 2) * ENT_DIM + d];
    }
  }
  __syncthreads();

  // ---------------- phase 3: out[16 x 256] = v[16 x 64] @ M[64 x 256] via f32 WMMA ----------------
  const int half = lane >> 4;     // 0: K-pair {0,1}; 1: K-pair {2,3} (ISA 16x4 A layout)
  const int ll = lane & 15;
  const int c2 = half * 2;
  const float2* MfragP = (const float2*)(ws + WS_MFRAG);
  const float* cvec = ws + WS_CVEC;

  for (int T = 0; T < 2; ++T) {
    // A fragments: lane<16 -> (M=ll, K=4kk+{0,1}); lane>=16 -> (M=ll, K=4kk+{2,3})
    v2f A[16];
#pragma unroll
    for (int kk = 0; kk < 16; ++kk) {
      float2 q = *(const float2*)&lds_v[(T * 16 + ll) * 64 + 4 * kk + c2];
      A[kk].x = q.x;
      A[kk].y = q.y;
    }
    for (int t = 0; t < 16; ++t) {
      v8f acc = {0.f, 0.f, 0.f, 0.f, 0.f, 0.f, 0.f, 0.f};
      const float2* bp = MfragP + (t * 16) * 32 + lane;
#pragma unroll
      for (int kk = 0; kk < 16; ++kk) {
        float2 bq = bp[kk * 32];
        v2f b;
        b.x = bq.x;
        b.y = bq.y;
        acc = __builtin_amdgcn_wmma_f32_16x16x4_f32(
            /*neg_a=*/false, A[kk], /*neg_b=*/false, b,
            /*c_mod=*/(short)0, acc, /*reuse_a=*/false, /*reuse_b=*/false);
      }
      float bias = cvec[t * 16 + ll];
      // D vgpr j: lanes<16 -> row j, lanes>=16 -> row j+8
      long long rowbase = (long long)(sample0 + T * 16 + half * 8) * HID + t * 16 + ll;
#pragma unroll
      for (int j = 0; j < 8; ++j) {
        out[rowbase + (long long)j * HID] = acc[j] + bias;
      }
    }
  }
}

extern "C" void kernel_launch(void* const* d_in, const int* in_sizes, int n_in,
                              void* d_out, int out_size, void* d_ws, size_t ws_size,
                              hipStream_t stream) {
  const float* x    = (const float*)d_in[0];
  const float* Watt = (const float*)d_in[1];
  // d_in[2] = batt: cancels inside softmax (uniform shift) -> unused
  const float* We   = (const float*)d_in[3];
  const float* be   = (const float*)d_in[4];
  const float* Wse  = (const float*)d_in[5];
  const float* bse  = (const float*)d_in[6];
  const float* Wg   = (const float*)d_in[7];
  const float* bg   = (const float*)d_in[8];
  const float* Wo   = (const float*)d_in[9];
  const float* bo   = (const float*)d_in[10];
  float* ws  = (float*)d_ws;
  float* out = (float*)d_out;

  fold_weights_kernel<<<(WS_TOTAL + 255) / 256, 256, 0, stream>>>(
      We, be, Wse, bse, Wg, bg, Wo, bo, ws);
  obs_embed_kernel<<<NSAMPLES / 32, 32, 0, stream>>>(x, Watt, ws, out);
}